// ComputeVecLoss_12506944766664
// MI455X (gfx1250) — compile-verified
//
#include <hip/hip_runtime.h>
#include <math.h>

typedef __attribute__((ext_vector_type(2))) float v2f;
typedef __attribute__((ext_vector_type(8))) float v8f;

#define NB      16      // batches
#define NP      17      // recon points per batch
#define NM      4096    // gt points per batch
#define NC      5       // interpolation points per segment
#define NQ      (NP * NP * NC)   // 1445 query points per batch
#define NTILES  ((NQ + 15) / 16) // 91 row tiles per batch
#define QPAD    (NTILES * 16)    // 1456 padded rows per batch
#define GTILES  (NM / 16)        // 256 gt tiles per batch
#define NWAVES  (NB * NTILES)    // 1456 waves total
#define WPB     4                // waves per block (128 threads)

// Workspace layout (d_ws):
//   Bpre:   NB * GTILES * 32 float2  (pre-formatted WMMA B operands, 1 MB)
//   minima: NB * QPAD float          (per-query min d2, 93 KB)
#define BPRE_ELEMS   (NB * GTILES * 32)          // float2 elements
#define MINIMA_OFF   (BPRE_ELEMS * 2)            // float offset into ws

// Kernel 0: pre-format the B matrices once per batch in WMMA lane layout.
// B col (gt point g) = [-2gx, -2gy, 1, g2]; 32-bit B 4x16 layout:
//   lanes 0-15 hold rows K0,K1 (col = lane), lanes 16-31 hold rows K2,K3.
__global__ void bprep_kernel(const float* __restrict__ gt,
                             float2* __restrict__ Bpre) {
    const int idx = blockIdx.x * blockDim.x + threadIdx.x;   // [0, NB*GTILES*32)
    if (idx >= BPRE_ELEMS) return;
    const int lane  = idx & 31;
    const int gtile = (idx >> 5) & (GTILES - 1);
    const int b     = idx >> 13;                              // / (GTILES*32)
    const int m     = gtile * 16 + (lane & 15);
    const float2 g  = ((const float2*)gt)[b * NM + m];
    float2 v;
    if (lane < 16) { v.x = -2.0f * g.x;  v.y = -2.0f * g.y; }          // K0,K1
    else           { v.x = 1.0f;         v.y = g.x * g.x + g.y * g.y; } // K2,K3
    Bpre[idx] = v;
}

// Kernel 1: per wave, 16 query points x 4096 gt points.
// d2 = ||k||^2 - 2 k.g + ||g||^2 folded into one f32 WMMA (16x16x4):
//   A row = [kx, ky, k2, 1] ; B col = [-2gx, -2gy, 1, g2]  =>  D = d2 tile.
__global__ void edge_min_kernel(const float* __restrict__ recon,
                                const float2* __restrict__ Bpre,
                                float* __restrict__ minima) {
    const int lane    = threadIdx.x & 31;
    const int waveBlk = threadIdx.x >> 5;
    const int waveId  = blockIdx.x * WPB + waveBlk;
    if (waveId >= NWAVES) return;           // wave-uniform: EXEC stays all-1s below

    const int b    = waveId / NTILES;
    const int tile = waveId % NTILES;
    const int half = lane >> 4;             // 0: lanes 0-15, 1: lanes 16-31
    const int r    = lane & 15;             // row (A) / column (B) index in tile

    // ---- Build this lane's slice of the A matrix (16x4 f32, 2 VGPRs/lane) ----
    // Query point q -> (i, j, c): K = p1[i]*t + p1[j]*(1-t), t = c/4.
    int q = tile * 16 + r;
    if (q > NQ - 1) q = NQ - 1;             // clamp padding rows (never stored)
    const int c  = q % NC;
    const int ij = q / NC;
    const int i  = ij / NP;
    const int j  = ij % NP;
    const float t = 0.25f * (float)c;

    const float2* p1 = (const float2*)recon + b * NP;
    const float2 pi = p1[i];
    const float2 pj = p1[j];
    const float kx = pi.x * t + pj.x * (1.0f - t);
    const float ky = pi.y * t + pj.y * (1.0f - t);
    const float k2 = kx * kx + ky * ky;

    // 32-bit A 16x4 layout: lanes 0-15 hold K0,K1 (row=lane); lanes 16-31 hold K2,K3.
    v2f A;
    A.x = half ? k2   : kx;
    A.y = half ? 1.0f : ky;

    v8f cz = {};
    v8f rmin;
#pragma unroll
    for (int e = 0; e < 8; ++e) rmin[e] = 3.0e38f;

    // Pre-formatted B stream for this batch: pure b64 load -> WMMA -> min.
    const v2f* bp = (const v2f*)Bpre + b * (GTILES * 32) + lane;

#pragma unroll 4
    for (int gtile = 0; gtile < GTILES; ++gtile) {
        const v2f Bm = bp[gtile * 32];

        v8f d = __builtin_amdgcn_wmma_f32_16x16x4_f32(
            /*neg_a=*/false, A, /*neg_b=*/false, Bm,
            /*c_mod=*/(short)0, cz, /*reuse_a=*/false, /*reuse_b=*/false);

#pragma unroll
        for (int e = 0; e < 8; ++e) rmin[e] = fminf(rmin[e], d[e]);
    }

    // ---- Min over columns: butterfly within each 16-lane half ----
    // C/D layout: lane group = column N, VGPR e = row (half 0: rows 0-7, half 1: rows 8-15).
#pragma unroll
    for (int m = 8; m >= 1; m >>= 1) {
#pragma unroll
        for (int e = 0; e < 8; ++e) {
            const float o = __shfl_xor(rmin[e], m, 32);
            rmin[e] = fminf(rmin[e], o);
        }
    }

    // Lane 0 holds rows 0..7, lane 16 holds rows 8..15.
    if (r == 0) {
        const int base = tile * 16 + half * 8;
#pragma unroll
        for (int e = 0; e < 8; ++e) {
            const int row = base + e;
            if (row < NQ) minima[b * QPAD + row] = rmin[e];
        }
    }
}

// Kernel 2: mask -> cosine sum -> scalar. 2176 (b, i>j) entries, one block.
__global__ void finalize_kernel(const float* __restrict__ recon,
                                const float* __restrict__ minima,
                                float* __restrict__ out) {
    const int tid = threadIdx.x;
    const int PAIRS = NP * (NP - 1) / 2;        // 136
    float cos_sum = 0.0f;
    float cnt     = 0.0f;
    const float2* p1 = (const float2*)recon;

    for (int e = tid; e < NB * PAIRS; e += 256) {
        const int b = e / PAIRS;
        int p = e % PAIRS;
        int i = 1, acc = 0;                      // map p -> (i, j) with i > j
        while (p >= acc + i) { acc += i; ++i; }
        const int j = p - acc;

        const float* mb = minima + b * QPAD + (i * NP + j) * NC;
        const float s = mb[0] + mb[1] + mb[2] + mb[3] + mb[4];
        if (s * 0.2f < 1e-3f) {                  // mean over C of min d2 < MAXDIS
            const float2 p0 = p1[b * NP + 0];
            const float2 pa = p1[b * NP + i];
            const float2 pb = p1[b * NP + j];
            const float uix = p0.x - pa.x, uiy = p0.y - pa.y;
            const float ujx = p0.x - pb.x, ujy = p0.y - pb.y;
            const float dot = fabsf(uix * ujx + uiy * ujy);
            const float ai  = sqrtf(uix * uix + uiy * uiy + 2e-5f); // +EPS per coord
            const float aj  = sqrtf(ujx * ujx + ujy * ujy + 2e-5f);
            cos_sum += dot / (ai * aj);
            cnt     += 1.0f;
        }
    }

    __shared__ float scos[256];
    __shared__ float scnt[256];
    scos[tid] = cos_sum;
    scnt[tid] = cnt;
    __syncthreads();
#pragma unroll
    for (int s = 128; s > 0; s >>= 1) {
        if (tid < s) { scos[tid] += scos[tid + s]; scnt[tid] += scnt[tid + s]; }
        __syncthreads();
    }
    if (tid == 0) out[0] = scos[0] / (1.0f + scnt[0]);
}

extern "C" void kernel_launch(void* const* d_in, const int* in_sizes, int n_in,
                              void* d_out, int out_size, void* d_ws, size_t ws_size,
                              hipStream_t stream) {
    const float* recon = (const float*)d_in[0];   // [16,17,2] f32
    const float* gt    = (const float*)d_in[1];   // [16,4096,2] f32
    float* out    = (float*)d_out;                // scalar f32
    float2* Bpre  = (float2*)d_ws;                // 1 MB pre-formatted B operands
    float* minima = (float*)d_ws + MINIMA_OFF;    // 93 KB per-query minima

    bprep_kernel<<<(BPRE_ELEMS + 255) / 256, 256, 0, stream>>>(gt, Bpre);

    const int grid1 = (NWAVES + WPB - 1) / WPB;   // 364 blocks x 128 threads
    edge_min_kernel<<<grid1, WPB * 32, 0, stream>>>(recon, Bpre, minima);

    finalize_kernel<<<1, 256, 0, stream>>>(recon, minima, out);
}